// TransferModelv2_51342039056577
// MI455X (gfx1250) — compile-verified
//
#include <hip/hip_runtime.h>
#include <hip/hip_bf16.h>

// Problem constants (from reference)
#define BB   16
#define LL   512
#define KNN  48
#define HD   128
#define CD   128
#define DIN  640   // 3*128 + 128 + 128
#define H1   512
#define H2   256

typedef float v2f __attribute__((ext_vector_type(2)));
typedef float v8f __attribute__((ext_vector_type(8)));

__device__ __forceinline__ float nan0(float v) { return (v != v) ? 0.0f : v; }

// -----------------------------------------------------------------------------
// Kernel 1: per-batch feature build.
// Collapses the full [L,L] distance + top-k to: for each mutation position,
// compute its distance row (512 dists), the row max (for masked adjustment),
// and the *rank* of the other mutation position (lax.top_k tie-break = lower
// index wins). rank < 48  <=>  "other mutation is a neighbor"; rank is the
// argmax-of-first-match index into the gathered edge row.
// Then builds emb[2,640], validity-weighted mean, LayerNorm, ReLU -> x1[b,640].
// -----------------------------------------------------------------------------
__global__ __launch_bounds__(256) void build_feats_kernel(
    const float* __restrict__ X,          // [B,L,4,3]
    const float* __restrict__ mask,       // [B,L]
    const int*   __restrict__ mut_pos,    // [B,2]
    const int*   __restrict__ mut_wt,     // [B,2]
    const int*   __restrict__ mut_mut,    // [B,2]
    const float* __restrict__ hid1,       // [B,L,128]
    const float* __restrict__ hid2,       // [B,L,128]
    const float* __restrict__ mpnn_embed, // [B,L,128]
    const float* __restrict__ mpnn_edges, // [B,L,48,128]
    const float* __restrict__ Ws,         // [21,128]
    const float* __restrict__ ln_g,       // [640]
    const float* __restrict__ ln_b,       // [640]
    float* __restrict__ x1)               // [B,640] = relu(LayerNorm(agg))
{
  __shared__ float Dsh[LL];
  __shared__ float redf[256];
  __shared__ int   redi[256];
  __shared__ float aggS[DIN];
  __shared__ int   sHas[2];
  __shared__ int   sIdx[2];

  const int b   = blockIdx.x;
  const int tid = threadIdx.x;

  int pos[2], wt[2], mt[2];
#pragma unroll
  for (int m = 0; m < 2; ++m) {
    pos[m] = mut_pos[b * 2 + m];
    wt[m]  = mut_wt [b * 2 + m];
    mt[m]  = mut_mut[b * 2 + m];
  }

  // ---- neighbor-rank computation for each mutation ----
  for (int m = 0; m < 2; ++m) {
    const int pm = pos[m];
    const int po = pos[1 - m];
    const float cx = nan0(X[((b * LL + pm) * 4 + 1) * 3 + 0]);
    const float cy = nan0(X[((b * LL + pm) * 4 + 1) * 3 + 1]);
    const float cz = nan0(X[((b * LL + pm) * 4 + 1) * 3 + 2]);
    const float mm = mask[b * LL + pm];

    for (int l = tid; l < LL; l += 256) {
      const float lx = nan0(X[((b * LL + l) * 4 + 1) * 3 + 0]);
      const float ly = nan0(X[((b * LL + l) * 4 + 1) * 3 + 1]);
      const float lz = nan0(X[((b * LL + l) * 4 + 1) * 3 + 2]);
      const float m2 = mm * mask[b * LL + l];
      const float dx = cx - lx, dy = cy - ly, dz = cz - lz;
      Dsh[l] = m2 * sqrtf(dx * dx + dy * dy + dz * dz + 1e-6f);
    }
    __syncthreads();

    // row max (for masked-out adjustment in reference)
    redf[tid] = fmaxf(Dsh[tid], Dsh[tid + 256]);
    __syncthreads();
    for (int s = 128; s > 0; s >>= 1) {
      if (tid < s) redf[tid] = fmaxf(redf[tid], redf[tid + s]);
      __syncthreads();
    }
    const float rowmax = redf[0];
    __syncthreads();

    // rank of pos_o in ascending-distance order, ties -> lower index first
    const float m2o   = mm * mask[b * LL + po];
    const float DadjO = Dsh[po] + (1.0f - m2o) * rowmax;
    int cnt = 0;
    for (int l = tid; l < LL; l += 256) {
      const float m2 = mm * mask[b * LL + l];
      const float da = Dsh[l] + (1.0f - m2) * rowmax;
      cnt += (da < DadjO) ? 1 : 0;
      cnt += (da == DadjO && l < po) ? 1 : 0;
    }
    redi[tid] = cnt;
    __syncthreads();
    for (int s = 128; s > 0; s >>= 1) {
      if (tid < s) redi[tid] += redi[tid + s];
      __syncthreads();
    }
    if (tid == 0) {
      const int rank = redi[0];
      sHas[m] = (rank < KNN) ? 1 : 0;
      sIdx[m] = rank;
    }
    __syncthreads();
  }

  // ---- emb build + validity-weighted mean over the 2 mutations ----
  const float v0 = ((mt[0] + wt[0] + pos[0]) != 0) ? 1.0f : 0.0f;
  const float v1 = ((mt[1] + wt[1] + pos[1]) != 0) ? 1.0f : 0.0f;
  const float invden = 1.0f / fmaxf(v0 + v1, 1.0f);
  const int has0 = sHas[0], has1 = sHas[1];
  const int idx0 = sIdx[0], idx1 = sIdx[1];

  for (int d = tid; d < DIN; d += 256) {
    float e0, e1;
    if (d < 128) {
      e0 = hid1[(b * LL + pos[0]) * HD + d];
      e1 = hid1[(b * LL + pos[1]) * HD + d];
    } else if (d < 256) {
      e0 = hid2[(b * LL + pos[0]) * HD + (d - 128)];
      e1 = hid2[(b * LL + pos[1]) * HD + (d - 128)];
    } else if (d < 384) {
      e0 = mpnn_embed[(b * LL + pos[0]) * HD + (d - 256)];
      e1 = mpnn_embed[(b * LL + pos[1]) * HD + (d - 256)];
    } else if (d < 512) {
      e0 = Ws[mt[0] * HD + (d - 384)];
      e1 = Ws[mt[1] * HD + (d - 384)];
    } else {
      e0 = has0 ? mpnn_edges[((b * LL + pos[0]) * KNN + idx0) * CD + (d - 512)] : 0.0f;
      e1 = has1 ? mpnn_edges[((b * LL + pos[1]) * KNN + idx1) * CD + (d - 512)] : 0.0f;
    }
    aggS[d] = (e0 * v0 + e1 * v1) * invden;
  }
  __syncthreads();

  // ---- LayerNorm (two-pass, matches mean((x-mu)^2)) + ReLU ----
  float s = 0.0f;
  for (int d = tid; d < DIN; d += 256) s += aggS[d];
  redf[tid] = s;
  __syncthreads();
  for (int st = 128; st > 0; st >>= 1) {
    if (tid < st) redf[tid] += redf[tid + st];
    __syncthreads();
  }
  const float mu = redf[0] * (1.0f / DIN);
  __syncthreads();

  float s2 = 0.0f;
  for (int d = tid; d < DIN; d += 256) {
    const float t = aggS[d] - mu;
    s2 += t * t;
  }
  redf[tid] = s2;
  __syncthreads();
  for (int st = 128; st > 0; st >>= 1) {
    if (tid < st) redf[tid] += redf[tid + st];
    __syncthreads();
  }
  const float var = redf[0] * (1.0f / DIN);
  const float inv = rsqrtf(var + 1e-5f);

  for (int d = tid; d < DIN; d += 256) {
    const float y = (aggS[d] - mu) * inv * ln_g[d] + ln_b[d];
    x1[b * DIN + d] = fmaxf(y, 0.0f);
  }
}

// -----------------------------------------------------------------------------
// fp32 WMMA GEMM, D[16,N] = relu?(A[16,K] @ B[K,N] + bias).
// One wave = one 16x16 output tile via V_WMMA_F32_16X16X4_F32, K stepped by 4.
// K/N are template constants so every A/B access is a global_load with an
// *immediate* offset (max 1.3MB < 24-bit range) -- no per-step address math.
// Double-buffered chunks of 8 k-steps (24 loads in flight) overlap the memory
// latency of chunk i+1 with the 8 WMMAs of chunk i.
// VGPR layouts per CDNA5 ISA 7.12.2:
//   A 16x4 f32: lanes 0-15 row=lane K={k,k+1}; lanes 16-31 row=lane-16 K={k+2,k+3}
//   B 4x16 f32: lanes 0-15 col=lane rows {k,k+1}; lanes 16-31 rows {k+2,k+3}
//   C/D 16x16:  VGPR v -> row v (lanes 0-15) / row v+8 (lanes 16-31), col=lane&15
// -----------------------------------------------------------------------------
#define CHUNK 8  // wmma steps per chunk; 32 K-values

template <int N, int CH>
__device__ __forceinline__ void load_frag(v2f* a, v2f* b,
                                          const float* __restrict__ ap,
                                          const float* __restrict__ bp) {
#pragma unroll
  for (int i = 0; i < CH; ++i) {
    a[i].x = ap[4 * i + 0];
    a[i].y = ap[4 * i + 1];
    b[i].x = bp[(size_t)(4 * i + 0) * N];
    b[i].y = bp[(size_t)(4 * i + 1) * N];
  }
}

template <int CH>
__device__ __forceinline__ void mma_frag(v8f& c, const v2f* a, const v2f* b) {
#pragma unroll
  for (int i = 0; i < CH; ++i) {
    c = __builtin_amdgcn_wmma_f32_16x16x4_f32(
        /*neg_a=*/false, a[i], /*neg_b=*/false, b[i],
        /*c_mod=*/(short)0, c, /*reuse_a=*/false, /*reuse_b=*/false);
  }
}

template <int K, int N, int DO_RELU>
__global__ __launch_bounds__(128) void mlp_wmma_kernel(
    const float* __restrict__ A,    // [16,K]
    const float* __restrict__ Bw,   // [K,N]
    const float* __restrict__ bias, // [N]
    float* __restrict__ out)        // [16,N]
{
  constexpr int NCH = K / (4 * CHUNK);  // 20 (layer1) / 16 (layer2)
  static_assert(K % (4 * CHUNK) == 0 && NCH % 2 == 0, "K must be 64-aligned");

  const int lane    = threadIdx.x & 31;
  const int wave    = threadIdx.x >> 5;
  const int tile    = blockIdx.x * (blockDim.x >> 5) + wave;
  const int colBase = tile * 16;
  const int half    = lane >> 4;  // 0: K pair {k,k+1}; 1: {k+2,k+3}
  const int lr      = lane & 15;

  const float* ap = A + lr * K + half * 2;
  const float* bp = Bw + (size_t)(half * 2) * N + colBase + lr;

  v8f c = {};
  v2f a0[CHUNK], b0[CHUNK], a1[CHUNK], b1[CHUNK];

  // prologue: chunk 0 -> buf0
  load_frag<N, CHUNK>(a0, b0, ap, bp);
  ap += 4 * CHUNK;
  bp += (size_t)(4 * CHUNK) * N;

  for (int t = 0; t < NCH - 2; t += 2) {
    load_frag<N, CHUNK>(a1, b1, ap, bp);   // chunk t+1
    ap += 4 * CHUNK;
    bp += (size_t)(4 * CHUNK) * N;
    mma_frag<CHUNK>(c, a0, b0);            // chunk t
    load_frag<N, CHUNK>(a0, b0, ap, bp);   // chunk t+2
    ap += 4 * CHUNK;
    bp += (size_t)(4 * CHUNK) * N;
    mma_frag<CHUNK>(c, a1, b1);            // chunk t+1
  }
  // epilogue: chunk NCH-2 is in buf0; load final chunk NCH-1, then drain
  load_frag<N, CHUNK>(a1, b1, ap, bp);
  mma_frag<CHUNK>(c, a0, b0);
  mma_frag<CHUNK>(c, a1, b1);

  const int col = colBase + lr;
  const float bv = bias[col];
#pragma unroll
  for (int v = 0; v < 8; ++v) {
    const int row = v + half * 8;
    float val = c[v] + bv;
    if (DO_RELU) val = fmaxf(val, 0.0f);
    out[row * N + col] = val;
  }
}

// -----------------------------------------------------------------------------
// Kernel 4: final 256 -> 1 projection per batch (no ReLU at output).
// -----------------------------------------------------------------------------
__global__ __launch_bounds__(256) void final_dot_kernel(
    const float* __restrict__ x3,  // [16,256] (already ReLU'd)
    const float* __restrict__ W3,  // [256,1]
    const float* __restrict__ b3,  // [1]
    float* __restrict__ out)       // [16]
{
  __shared__ float red[256];
  const int b = blockIdx.x, tid = threadIdx.x;
  red[tid] = x3[b * H2 + tid] * W3[tid];
  __syncthreads();
  for (int s = 128; s > 0; s >>= 1) {
    if (tid < s) red[tid] += red[tid + s];
    __syncthreads();
  }
  if (tid == 0) out[b] = red[0] + b3[0];
}

extern "C" void kernel_launch(void* const* d_in, const int* in_sizes, int n_in,
                              void* d_out, int out_size, void* d_ws, size_t ws_size,
                              hipStream_t stream) {
  const float* X          = (const float*)d_in[0];
  // d_in[1] = S (dead: the wildtype write-back never feeds the output)
  const float* mask       = (const float*)d_in[2];
  // d_in[3..5] chain_M / residue_idx / chain_encoding_all: unused by output
  const int*   mut_pos    = (const int*)d_in[6];
  const int*   mut_wt     = (const int*)d_in[7];
  const int*   mut_mut    = (const int*)d_in[8];
  // d_in[9] mut_ddGs, d_in[10] atom_mask: unused
  const float* hid1       = (const float*)d_in[11];
  const float* hid2       = (const float*)d_in[12];
  const float* mpnn_embed = (const float*)d_in[13];
  const float* mpnn_edges = (const float*)d_in[14];
  const float* Ws         = (const float*)d_in[15];
  const float* ln_g       = (const float*)d_in[16];
  const float* ln_b       = (const float*)d_in[17];
  const float* W1         = (const float*)d_in[18];
  const float* b1         = (const float*)d_in[19];
  const float* W2         = (const float*)d_in[20];
  const float* b2         = (const float*)d_in[21];
  const float* W3         = (const float*)d_in[22];
  const float* b3         = (const float*)d_in[23];
  float* out = (float*)d_out;

  float* x1 = (float*)d_ws;          // [16,640]
  float* x2 = x1 + BB * DIN;         // [16,512]
  float* x3 = x2 + BB * H1;          // [16,256]

  build_feats_kernel<<<BB, 256, 0, stream>>>(
      X, mask, mut_pos, mut_wt, mut_mut, hid1, hid2, mpnn_embed, mpnn_edges,
      Ws, ln_g, ln_b, x1);

  // layer 1: [16,640]@[640,512] -> 32 tiles, 4 waves/block -> 8 blocks
  mlp_wmma_kernel<DIN, H1, 1><<<(H1 / 16) / 4, 128, 0, stream>>>(x1, W1, b1, x2);
  // layer 2: [16,512]@[512,256] -> 16 tiles -> 4 blocks
  mlp_wmma_kernel<H1, H2, 1><<<(H2 / 16) / 4, 128, 0, stream>>>(x2, W2, b2, x3);
  // layer 3: [16,256]@[256,1]
  final_dot_kernel<<<BB, 256, 0, stream>>>(x3, W3, b3, out);
}